// SNN_2405181685794
// MI455X (gfx1250) — compile-verified
//
#include <hip/hip_runtime.h>

typedef float v2f __attribute__((ext_vector_type(2)));
typedef float v8f __attribute__((ext_vector_type(8)));

#define T_STEPS 64
#define BATCH   2048
#define H1      1024
#define KIN     784
#define H2      10

// Fused SNN kernel: one block per 16-row batch tile, 8 waves (wave32) per block.
// Phase 1: c1 = x @ W1^T via V_WMMA_F32_16X16X4_F32 (exact fp32), kept in VGPRs.
// Phase 2: 64 timesteps of LIF dynamics + s1 @ W2^T via WMMA (s1 bounced
//          through LDS to convert D-layout -> A-layout), cross-wave partial
//          reduction in LDS, wave 0 owns v2 state and writes spikes.
__global__ __launch_bounds__(256) void snn_fused(const float* __restrict__ x,
                                                 const float* __restrict__ W1,
                                                 const float* __restrict__ W2,
                                                 float* __restrict__ out) {
    __shared__ float s1lds[16 * H1];    // 64 KB: spike matrix tile, row-major [16][1024]
    __shared__ float w2lds[16 * H1];    // 64 KB: W2 zero-padded to 16 rows, row-major
    __shared__ float c2part[8 * 256];   //  8 KB: per-wave partial c2 tiles [8][16][16]

    const int tid  = threadIdx.x;
    const int w    = tid >> 5;          // wave id 0..7
    const int lane = tid & 31;
    const int hl   = lane & 15;         // lane % 16
    const int hi   = lane >> 4;         // 0 for lanes 0-15, 1 for lanes 16-31
    const int r0   = blockIdx.x * 16;   // batch row base of this block

    // Stage W2 into LDS, zero-padding rows 10..15 so GEMM2 cols 10..15 produce 0.
    for (int idx = tid; idx < 16 * H1; idx += 256) {
        const int row = idx >> 10;
        const int col = idx & (H1 - 1);
        w2lds[idx] = (row < H2) ? W2[row * H1 + col] : 0.0f;
    }
    __syncthreads();

    // ---------------- Phase 1: c1 tile (16 x 128 per wave) ----------------
    v8f acc[8];
#pragma unroll
    for (int i = 0; i < 8; ++i) acc[i] = v8f{0.f, 0.f, 0.f, 0.f, 0.f, 0.f, 0.f, 0.f};

    const float* xrow = x + (size_t)(r0 + hl) * KIN;   // A: row m = hl of the batch tile
    for (int ks = 0; ks < KIN / 4; ++ks) {
        const int kb = ks * 4 + hi * 2;                // K pair for this lane half
        const v2f a = *(const v2f*)(xrow + kb);        // A[m][kb], A[m][kb+1]
#pragma unroll
        for (int i = 0; i < 8; ++i) {
            const int n = (w * 8 + i) * 16 + hl;       // B col n -> W1 row n
            const v2f b = *(const v2f*)(W1 + (size_t)n * KIN + kb); // W1[n][kb..kb+1]
            acc[i] = __builtin_amdgcn_wmma_f32_16x16x4_f32(
                false, a, false, b, (short)0, acc[i], false, false);
        }
    }

    // ---------------- Phase 2: 64 timesteps ----------------
    v8f v1[8];
#pragma unroll
    for (int i = 0; i < 8; ++i) v1[i] = v8f{0.f, 0.f, 0.f, 0.f, 0.f, 0.f, 0.f, 0.f};
    v8f v2 = v8f{0.f, 0.f, 0.f, 0.f, 0.f, 0.f, 0.f, 0.f};

    for (int t = 0; t < T_STEPS; ++t) {
        // LIF layer 1: v = 0.5*(v + c1); spike if v >= 1; hard reset to 0.
#pragma unroll
        for (int i = 0; i < 8; ++i) {
            const v8f vn  = 0.5f * (v1[i] + acc[i]);
            const int col = (w * 8 + i) * 16 + hl;
#pragma unroll
            for (int e = 0; e < 8; ++e) {
                const float vv    = vn[e];
                const bool  fired = (vv >= 1.0f);
                v1[i][e] = fired ? 0.0f : vv;
                const int row = e + hi * 8;            // C/D layout row
                s1lds[row * H1 + col] = fired ? 1.0f : 0.0f;
            }
        }
        __syncthreads();

        // GEMM2 K-slice: wave w covers k in [w*128, w*128+128).
        v8f c2 = v8f{0.f, 0.f, 0.f, 0.f, 0.f, 0.f, 0.f, 0.f};
#pragma unroll 4
        for (int kk = 0; kk < 32; ++kk) {
            const int kb = w * 128 + kk * 4 + hi * 2;
            const v2f a = *(const v2f*)(s1lds + hl * H1 + kb);  // A = s1[m][k]
            const v2f b = *(const v2f*)(w2lds + hl * H1 + kb);  // B[k][n] = W2pad[n][k]
            c2 = __builtin_amdgcn_wmma_f32_16x16x4_f32(
                false, a, false, b, (short)0, c2, false, false);
        }
#pragma unroll
        for (int e = 0; e < 8; ++e) {
            c2part[w * 256 + (e + hi * 8) * 16 + hl] = c2[e];
        }
        __syncthreads();

        // Wave 0: reduce partials, LIF layer 2, emit spikes for this timestep.
        if (w == 0) {
#pragma unroll
            for (int e = 0; e < 8; ++e) {
                const int row = e + hi * 8;
                float sum = 0.0f;
#pragma unroll
                for (int ww = 0; ww < 8; ++ww)
                    sum += c2part[ww * 256 + row * 16 + hl];
                const float vv    = 0.5f * (v2[e] + sum);
                const bool  fired = (vv >= 1.0f);
                v2[e] = fired ? 0.0f : vv;
                if (hl < H2) {
                    out[((size_t)t * BATCH + (r0 + row)) * H2 + hl] = fired ? 1.0f : 0.0f;
                }
            }
        }
    }
}

extern "C" void kernel_launch(void* const* d_in, const int* in_sizes, int n_in,
                              void* d_out, int out_size, void* d_ws, size_t ws_size,
                              hipStream_t stream) {
    const float* x  = (const float*)d_in[0];   // [2048, 28, 28] -> [2048, 784]
    const float* W1 = (const float*)d_in[1];   // [1024, 784]
    const float* W2 = (const float*)d_in[2];   // [10, 1024]
    float* out = (float*)d_out;                // [64, 2048, 10]

    dim3 grid(BATCH / 16);   // 128 blocks
    dim3 block(256);         // 8 wave32 waves
    hipLaunchKernelGGL(snn_fused, grid, block, 0, stream, x, W1, W2, out);
}